// NoSAF_32280974197073
// MI455X (gfx1250) — compile-verified
//
#include <hip/hip_runtime.h>
#include <math.h>

#define LRELU_NS 0.2f
#define BN_EPS   1e-5f

typedef float v2f __attribute__((ext_vector_type(2)));
typedef float v8f __attribute__((ext_vector_type(8)));
typedef unsigned int v4u __attribute__((ext_vector_type(4)));
typedef int v4i __attribute__((ext_vector_type(4)));
typedef int v8i __attribute__((ext_vector_type(8)));

// ---------------------------------------------------------------------------
// Tensor Data Mover: stage a 2-D fp32 tile (tile_rows x tile_cols, row stride
// row_stride elements) from global memory into LDS. D# layout per CDNA5 ISA
// (CDNA5 ISA). This toolchain's builtin takes 6 args.
// ---------------------------------------------------------------------------
__device__ inline void tdm_load_2d_f32(const float* gsrc, unsigned lds_off,
                                       int tile_cols, int tile_rows,
                                       int row_stride) {
  unsigned long long ga = (unsigned long long)(uintptr_t)gsrc;
  v4u g0;
  g0[0] = 1u;                                   // count=1 (valid user D#)
  g0[1] = lds_off;                              // lds_addr
  g0[2] = (unsigned)(ga & 0xFFFFFFFFu);         // global_addr[31:0]
  g0[3] = (unsigned)((ga >> 32) & 0x01FFFFFFu)  // global_addr[56:32]
          | (2u << 30);                         // type=2 ("image")
  v8i g1;
  g1[0] = (int)(2u << 16);                      // data_size=2 -> 4 bytes
  g1[1] = (int)(((unsigned)tile_cols & 0xFFFFu) << 16);        // tensor_dim0 lo
  g1[2] = (int)((((unsigned)tile_cols >> 16) & 0xFFFFu)        // tensor_dim0 hi
          | (((unsigned)tile_rows & 0xFFFFu) << 16));          // tensor_dim1 lo
  g1[3] = (int)((((unsigned)tile_rows >> 16) & 0xFFFFu)        // tensor_dim1 hi
          | (((unsigned)tile_cols & 0xFFFFu) << 16));          // tile_dim0
  g1[4] = (int)((unsigned)tile_rows & 0xFFFFu);                // tile_dim1 (tile_dim2=0)
  g1[5] = (int)row_stride;                      // tensor_dim0_stride[31:0]
  g1[6] = 0;
  g1[7] = 0;
  v4i z4 = {0, 0, 0, 0};
  v8i z8 = {0, 0, 0, 0, 0, 0, 0, 0};
  __builtin_amdgcn_tensor_load_to_lds(g0, g1, z4, z4, z8, 0);
}

// ---------------------------------------------------------------------------
// Degree / symmetric-norm kernels
// ---------------------------------------------------------------------------
__global__ void deg_init_kernel(float* deg, int N) {
  int i = blockIdx.x * blockDim.x + threadIdx.x;
  if (i < N) deg[i] = 1.0f;  // self loop contributes 1 to every node
}

__global__ void deg_count_kernel(const int* __restrict__ dst, float* deg, int E) {
  int e = blockIdx.x * blockDim.x + threadIdx.x;
  if (e < E) atomicAdd(&deg[dst[e]], 1.0f);
}

__global__ void deg_finalize_kernel(float* deg, int N) {
  int i = blockIdx.x * blockDim.x + threadIdx.x;
  if (i < N) deg[i] = rsqrtf(deg[i]);  // deg >= 1 always (self loops)
}

// ---------------------------------------------------------------------------
// fp32 WMMA GEMM with TDM-staged weights:
//   C[N,Dout] = act(A[N,K] @ W[K,Dout] + bias)
// Block = 256 threads = 8 wave32s covering 128 rows x one 64-column band.
// Wave 0 stages W[K x 64] into LDS via tensor_load_to_lds; the block then
// re-swizzles it to [K/4][64][4] so every WMMA B fragment (two K-adjacent
// values of one column) is a single aligned ds_load_b64. Each wave computes
// a 16x64 strip: one global b64 A-fragment load feeds 4 back-to-back
// V_WMMA_F32_16X16X4_F32 into 4 independent accumulators.
// ---------------------------------------------------------------------------
template <int K>
__global__ void gemm_wmma_lds_kernel(const float* __restrict__ A,
                                     const float* __restrict__ W,
                                     const float* __restrict__ bias,
                                     float* __restrict__ Cout,
                                     int Nrows, int Dout, int act) {
  __shared__ float lw[K * 64];   // raw band (TDM destination)
  __shared__ float lsw[K * 64];  // swizzled [K/4][64][4]

  const int wave = threadIdx.x >> 5;
  const int lane = threadIdx.x & 31;
  const int c0 = blockIdx.y << 6;  // column band start

  if (threadIdx.x < 32) {
    tdm_load_2d_f32(W + c0, (unsigned)(uintptr_t)&lw[0], 64, K, Dout);
    __builtin_amdgcn_s_wait_tensorcnt(0);
  }
  __syncthreads();

  // swizzle: lsw[((r/4)*64 + c)*4 + (r%4)] = lw[r*64 + c]
  for (int i = threadIdx.x; i < K * 64; i += 256) {
    int r = i >> 6;
    int c = i & 63;
    lsw[(((r >> 2) << 6) + c) * 4 + (r & 3)] = lw[i];
  }
  __syncthreads();

  const int m0 = (blockIdx.x * 8 + wave) << 4;
  if (m0 >= Nrows) return;  // after all barriers: safe

  // A fragment: lane<16 -> row m0+lane, K=k..k+1 ; lane>=16 -> K=k+2..k+3
  const int arow = m0 + (lane & 15);
  const int koff = (lane >> 4) << 1;  // 0 or 2
  const int col = lane & 15;

  const float* Aptr = A + (size_t)arow * K + koff;
  const float* Lb = &lsw[(col << 2) + koff];  // + k*64 per k-step

  v8f acc0 = {}, acc1 = {}, acc2 = {}, acc3 = {};
  for (int k = 0; k < K; k += 4) {
    v2f a = *(const v2f*)(Aptr + k);       // 8B aligned
    const float* lb = Lb + (k << 6);       // (k/4)*256 floats
    v2f b0 = *(const v2f*)(lb);            // col,      K=k+koff..k+koff+1
    v2f b1 = *(const v2f*)(lb + 64);       // col+16
    v2f b2 = *(const v2f*)(lb + 128);      // col+32
    v2f b3 = *(const v2f*)(lb + 192);      // col+48
    acc0 = __builtin_amdgcn_wmma_f32_16x16x4_f32(false, a, false, b0, (short)0, acc0, false, false);
    acc1 = __builtin_amdgcn_wmma_f32_16x16x4_f32(false, a, false, b1, (short)0, acc1, false, false);
    acc2 = __builtin_amdgcn_wmma_f32_16x16x4_f32(false, a, false, b2, (short)0, acc2, false, false);
    acc3 = __builtin_amdgcn_wmma_f32_16x16x4_f32(false, a, false, b3, (short)0, acc3, false, false);
  }

  float bv0 = 0.f, bv1 = 0.f, bv2 = 0.f, bv3 = 0.f;
  if (bias) {
    bv0 = bias[c0 + col];
    bv1 = bias[c0 + 16 + col];
    bv2 = bias[c0 + 32 + col];
    bv3 = bias[c0 + 48 + col];
  }
  const int mbase = m0 + ((lane >> 4) << 3);  // lanes 16-31 -> rows +8..+15
#pragma unroll
  for (int r = 0; r < 8; ++r) {
    float* rp = Cout + (size_t)(mbase + r) * Dout + c0 + col;
    float v0 = acc0[r] + bv0;
    float v1 = acc1[r] + bv1;
    float v2 = acc2[r] + bv2;
    float v3 = acc3[r] + bv3;
    if (act == 1) {
      v0 = fmaxf(v0, 0.f); v1 = fmaxf(v1, 0.f);
      v2 = fmaxf(v2, 0.f); v3 = fmaxf(v3, 0.f);
    }
    rp[0] = v0; rp[16] = v1; rp[32] = v2; rp[48] = v3;
  }
}

// ---------------------------------------------------------------------------
// BatchNorm column statistics (sum, sumsq) -> stats[0..C-1], stats[C..2C-1]
// ---------------------------------------------------------------------------
__global__ void zero_kernel(float* p, int n) {
  int i = blockIdx.x * blockDim.x + threadIdx.x;
  if (i < n) p[i] = 0.0f;
}

__global__ void bn_reduce_kernel(const float* __restrict__ x,
                                 float* __restrict__ stats,
                                 int total, int C) {
  __shared__ float ss[128];
  __shared__ float sq[128];
  for (int i = threadIdx.x; i < C; i += blockDim.x) { ss[i] = 0.0f; sq[i] = 0.0f; }
  __syncthreads();
  int cmask = C - 1;  // C is a power of two (64 or 128)
  for (int idx = blockIdx.x * blockDim.x + threadIdx.x; idx < total;
       idx += gridDim.x * blockDim.x) {
    float v = x[idx];
    int c = idx & cmask;
    atomicAdd(&ss[c], v);
    atomicAdd(&sq[c], v * v);
  }
  __syncthreads();
  for (int i = threadIdx.x; i < C; i += blockDim.x) {
    atomicAdd(&stats[i], ss[i]);
    atomicAdd(&stats[C + i], sq[i]);
  }
}

__global__ void bn_apply_relu_kernel(const float* __restrict__ x,
                                     float* __restrict__ y,
                                     const float* __restrict__ stats,
                                     const float* __restrict__ g,
                                     const float* __restrict__ be,
                                     int total, int C, float invN) {
  int idx = blockIdx.x * blockDim.x + threadIdx.x;
  if (idx >= total) return;
  int c = idx & (C - 1);
  float mu  = stats[c] * invN;
  float var = stats[C + c] * invN - mu * mu;
  float v = (x[idx] - mu) * rsqrtf(var + BN_EPS) * g[c] + be[c];
  y[idx] = fmaxf(v, 0.0f);
}

// ---------------------------------------------------------------------------
// GCN aggregation: agg[n,f] = bb[f] + m[n,f]*dinv[n]^2  (self loop + bias)
// then per-edge scatter agg[dst] += m[src] * dinv[src]*dinv[dst]
// ---------------------------------------------------------------------------
__global__ void agg_init_kernel(const float* __restrict__ m,
                                const float* __restrict__ dinv,
                                const float* __restrict__ bbias,
                                float* __restrict__ agg, int total) {
  int idx = blockIdx.x * blockDim.x + threadIdx.x;
  if (idx >= total) return;
  int node = idx >> 6;
  int f = idx & 63;
  float di = dinv[node];
  agg[idx] = bbias[f] + m[idx] * di * di;
}

// one wave per edge, 2 features per lane (D=64)
__global__ void scatter_kernel(const float* __restrict__ m,
                               const int* __restrict__ src,
                               const int* __restrict__ dst,
                               const float* __restrict__ dinv,
                               float* __restrict__ agg, int E) {
  int e = (blockIdx.x * blockDim.x + threadIdx.x) >> 5;
  int lane = threadIdx.x & 31;
  if (e >= E) return;
  int s = src[e];
  int d = dst[e];
  float w = dinv[s] * dinv[d];
  size_t sb = (size_t)s * 64;
  size_t db = (size_t)d * 64;
  float v0 = m[sb + lane]      * w;
  float v1 = m[sb + lane + 32] * w;
  atomicAdd(&agg[db + lane],      v0);
  atomicAdd(&agg[db + lane + 32], v1);
}

// ---------------------------------------------------------------------------
// Node-weight learner + residual fuse. One wave32 per node, D=64 (2 feats/lane).
//   t = first ? h : (h - fused)
//   z = leaky_relu(t @ lw1 + lb1);  w = sigmoid(z @ lw2 + lb2)
//   h *= w;  fused = first ? h : fused + h
// ---------------------------------------------------------------------------
__global__ void learner_fuse_kernel(float* __restrict__ h,
                                    float* __restrict__ fused,
                                    const float* __restrict__ lw1,
                                    const float* __restrict__ lb1,
                                    const float* __restrict__ lw2,
                                    const float* __restrict__ lb2,
                                    int N, int first) {
  int node = (blockIdx.x * blockDim.x + threadIdx.x) >> 5;
  int lane = threadIdx.x & 31;
  if (node >= N) return;
  size_t base = (size_t)node * 64;
  float h0 = h[base + lane];
  float h1 = h[base + lane + 32];
  float f0 = 0.0f, f1 = 0.0f;
  if (!first) { f0 = fused[base + lane]; f1 = fused[base + lane + 32]; }
  float t0 = h0 - f0;
  float t1 = h1 - f1;

  float s = lb2[0];
#pragma unroll
  for (int j = 0; j < 4; ++j) {
    float p = t0 * lw1[lane * 4 + j] + t1 * lw1[(lane + 32) * 4 + j];
#pragma unroll
    for (int off = 16; off > 0; off >>= 1) p += __shfl_xor(p, off, 32);
    float z = p + lb1[j];
    z = (z > 0.0f) ? z : LRELU_NS * z;
    s += z * lw2[j];
  }
  float w = 1.0f / (1.0f + __expf(-s));
  h0 *= w;
  h1 *= w;
  h[base + lane]      = h0;
  h[base + lane + 32] = h1;
  if (first) {
    fused[base + lane]      = h0;
    fused[base + lane + 32] = h1;
  } else {
    fused[base + lane]      = f0 + h0;
    fused[base + lane + 32] = f1 + h1;
  }
}

// ---------------------------------------------------------------------------
// Host launcher
// ---------------------------------------------------------------------------
static inline int cdiv(int a, int b) { return (a + b - 1) / b; }

extern "C" void kernel_launch(void* const* d_in, const int* in_sizes, int n_in,
                              void* d_out, int out_size, void* d_ws, size_t ws_size,
                              hipStream_t stream) {
  const float* x      = (const float*)d_in[0];
  const int*   ei     = (const int*)  d_in[1];
  const float* im_w1  = (const float*)d_in[2];
  const float* im_b1  = (const float*)d_in[3];
  const float* im_g1  = (const float*)d_in[4];
  const float* im_be1 = (const float*)d_in[5];
  const float* im_w2  = (const float*)d_in[6];
  const float* im_b2  = (const float*)d_in[7];
  const float* lw1    = (const float*)d_in[8];   // [9,64,4]
  const float* lb1    = (const float*)d_in[9];   // [9,4]
  const float* lw2    = (const float*)d_in[10];  // [9,4,1]
  const float* lb2    = (const float*)d_in[11];  // [9,1]
  const float* bw     = (const float*)d_in[12];  // [8,64,64]
  const float* bb     = (const float*)d_in[13];  // [8,64]
  const float* bg     = (const float*)d_in[14];  // [8,64]
  const float* bbe    = (const float*)d_in[15];  // [8,64]
  const float* om_w1  = (const float*)d_in[16];  // [64,128]
  const float* om_b1  = (const float*)d_in[17];
  const float* om_w2  = (const float*)d_in[18];  // [128,64]
  const float* om_b2  = (const float*)d_in[19];
  float* out = (float*)d_out;

  const int N = in_sizes[0] / 128;  // 50000
  const int E = in_sizes[1] / 2;    // 800000
  const int* src = ei;
  const int* dst = ei + E;

  // workspace layout (floats)
  float* ws    = (float*)d_ws;
  float* buf1  = ws;                          // [N,128] wide scratch
  float* h     = buf1  + (size_t)N * 128;     // [N,64]
  float* fused = h     + (size_t)N * 64;      // [N,64]
  float* m     = fused + (size_t)N * 64;      // [N,64]
  float* agg   = m     + (size_t)N * 64;      // [N,64]
  float* dinv  = agg   + (size_t)N * 64;      // [N]
  float* stats = dinv  + N;                   // [256]

  const int TB = 256;          // 8 wave32s per block
  const int RED_BLOCKS = 512;  // bn reduction grid
  const float invN = 1.0f / (float)N;
  const int mblocks = cdiv(N, 128);  // 128 rows per block

  // --- symmetric GCN normalization weights ---
  deg_init_kernel<<<cdiv(N, TB), TB, 0, stream>>>(dinv, N);
  deg_count_kernel<<<cdiv(E, TB), TB, 0, stream>>>(dst, dinv, E);
  deg_finalize_kernel<<<cdiv(N, TB), TB, 0, stream>>>(dinv, N);

  // --- input map: relu(bn(x@im_w1 + b1)) @ im_w2 + b2 ---
  gemm_wmma_lds_kernel<128><<<dim3(mblocks, 2), TB, 0, stream>>>(
      x, im_w1, im_b1, buf1, N, 128, 0);
  zero_kernel<<<1, 256, 0, stream>>>(stats, 256);
  bn_reduce_kernel<<<RED_BLOCKS, TB, 0, stream>>>(buf1, stats, N * 128, 128);
  bn_apply_relu_kernel<<<cdiv(N * 128, TB), TB, 0, stream>>>(
      buf1, buf1, stats, im_g1, im_be1, N * 128, 128, invN);
  gemm_wmma_lds_kernel<128><<<dim3(mblocks, 1), TB, 0, stream>>>(
      buf1, im_w2, im_b2, h, N, 64, 0);

  // --- learner 0: h *= sigmoid-mlp(h); fused = h ---
  learner_fuse_kernel<<<cdiv(N * 32, TB), TB, 0, stream>>>(
      h, fused, lw1, lb1, lw2, lb2, N, 1);

  // --- 8 GCN blocks ---
  for (int i = 0; i < 8; ++i) {
    gemm_wmma_lds_kernel<64><<<dim3(mblocks, 1), TB, 0, stream>>>(
        h, bw + (size_t)i * 64 * 64, nullptr, m, N, 64, 0);

    agg_init_kernel<<<cdiv(N * 64, TB), TB, 0, stream>>>(
        m, dinv, bb + (size_t)i * 64, agg, N * 64);
    scatter_kernel<<<cdiv(E * 32, TB), TB, 0, stream>>>(
        m, src, dst, dinv, agg, E);

    zero_kernel<<<1, 256, 0, stream>>>(stats, 128);
    bn_reduce_kernel<<<RED_BLOCKS, TB, 0, stream>>>(agg, stats, N * 64, 64);
    bn_apply_relu_kernel<<<cdiv(N * 64, TB), TB, 0, stream>>>(
        agg, h, stats, bg + (size_t)i * 64, bbe + (size_t)i * 64, N * 64, 64, invN);

    learner_fuse_kernel<<<cdiv(N * 32, TB), TB, 0, stream>>>(
        h, fused, lw1 + (size_t)(i + 1) * 256, lb1 + (size_t)(i + 1) * 4,
        lw2 + (size_t)(i + 1) * 4, lb2 + (i + 1), N, 0);
  }

  // --- output map: relu(fused@om_w1 + b1) @ om_w2 + b2 ---
  gemm_wmma_lds_kernel<64><<<dim3(mblocks, 2), TB, 0, stream>>>(
      fused, om_w1, om_b1, buf1, N, 128, 1);
  gemm_wmma_lds_kernel<128><<<dim3(mblocks, 1), TB, 0, stream>>>(
      buf1, om_w2, om_b2, out, N, 64, 0);
}